// Sto_Enc_21827023798807
// MI455X (gfx1250) — compile-verified
//
#include <hip/hip_runtime.h>
#include <math.h>

#define NN 100000
#define EE 800000
#define BB 4096
#define VV 103
#define DIN 200
#define HH 256

typedef __attribute__((ext_vector_type(2))) float v2f;
typedef __attribute__((ext_vector_type(8))) float v8f;

// ---------------------------------------------------------------------------
// WMMA f32 GEMM: out[M,Nout] = A[M,K] @ B (+bias, +accum, act)
// TRANSB==false: B stored [K,Nout] row-major ; TRANSB==true: B stored [Nout,K]
// Block = 256 threads = 8 waves; each wave owns one 16x16 tile.
// n-tile = (blockIdx.y*8 + waveId). Requires M%16==0, K%4==0, Nout%128==0.
// ACT: 0=none, 1=relu, 2=leaky(0.01). ACCUM: out += result.
// ---------------------------------------------------------------------------
template <bool TRANSB, int ACT, bool ACCUM>
__global__ __launch_bounds__(256)
void wmma_gemm(const float* __restrict__ A, const float* __restrict__ Bm,
               const float* __restrict__ bias, float* __restrict__ out,
               int M, int K, int Nout)
{
    const int wave = threadIdx.x >> 5;
    const int lane = threadIdx.x & 31;
    const int half = lane >> 4;       // 0: K=k0,k0+1   1: K=k0+2,k0+3
    const int lr   = lane & 15;
    const int m0   = blockIdx.x * 16;
    const int n0   = (blockIdx.y * 8 + wave) * 16;

    v8f acc = {0.f,0.f,0.f,0.f,0.f,0.f,0.f,0.f};

    // A fragment: lane holds A[m0+lr][k0+2*half .. +1]  (8B aligned)
    const float* ap = A + (size_t)(m0 + lr) * K + 2 * half;

    if (TRANSB) {
        // B^T fragment: lane holds W[n0+lr][k0+2*half .. +1]
        const float* bp = Bm + (size_t)(n0 + lr) * K + 2 * half;
#pragma unroll 4
        for (int k0 = 0; k0 < K; k0 += 4) {
            v2f af = *(const v2f*)ap;
            v2f bf = *(const v2f*)bp;
            acc = __builtin_amdgcn_wmma_f32_16x16x4_f32(
                      false, af, false, bf, (short)0, acc, false, false);
            ap += 4; bp += 4;
        }
    } else {
        // B fragment: lane holds B[k0+2*half][n0+lr], B[k0+2*half+1][n0+lr]
        const float* bp = Bm + (size_t)(2 * half) * Nout + n0 + lr;
        const size_t strideN = (size_t)Nout;
#pragma unroll 4
        for (int k0 = 0; k0 < K; k0 += 4) {
            v2f af = *(const v2f*)ap;
            v2f bf;
            bf.x = bp[0];
            bf.y = bp[strideN];
            acc = __builtin_amdgcn_wmma_f32_16x16x4_f32(
                      false, af, false, bf, (short)0, acc, false, false);
            ap += 4; bp += 4 * strideN;
        }
    }

    const float bv = bias ? bias[n0 + lr] : 0.f;
#pragma unroll
    for (int g = 0; g < 8; ++g) {
        const int row = m0 + g + 8 * half;
        const size_t o = (size_t)row * Nout + n0 + lr;
        float v = acc[g];
        if (ACCUM) v += out[o];
        v += bv;
        if (ACT == 1)      v = v > 0.f ? v : 0.f;
        else if (ACT == 2) v = v > 0.f ? v : 0.01f * v;
        out[o] = v;
    }
}

// ---------------------------------------------------------------------------
// small helpers / pointwise kernels
// ---------------------------------------------------------------------------
__global__ void fill_kernel(float* p, float v, int n) {
    int i = blockIdx.x * blockDim.x + threadIdx.x;
    if (i < n) p[i] = v;
}

__global__ void embed_kernel(const float* __restrict__ w, const int* __restrict__ sx,
                             float* __restrict__ xe, int total) {
    int i = blockIdx.x * blockDim.x + threadIdx.x;
    if (i >= total) return;
    int n = i / DIN, d = i - n * DIN;
    int ix = sx[n] - 1; if (ix < 0) ix += VV;   // torch-style wrap of index -1
    xe[i] = w[(size_t)ix * DIN + d];
}

__global__ void deg_kernel(const int* __restrict__ dst, float* deg, int e) {
    int i = blockIdx.x * blockDim.x + threadIdx.x;
    if (i < e) atomicAdd(&deg[dst[i]], 1.0f);
}

__global__ void dinv_kernel(float* deg, int n) {
    int i = blockIdx.x * blockDim.x + threadIdx.x;
    if (i < n) deg[i] = rsqrtf(fmaxf(deg[i], 1.0f));
}

// out[n,:] = hw[n,:] * dinv[n]^2   (self-loop contribution, single writer)
__global__ __launch_bounds__(HH)
void self_init_kernel(const float* __restrict__ hw, const float* __restrict__ dinv,
                      float* __restrict__ out) {
    int n = blockIdx.x, t = threadIdx.x;
    float d = dinv[n];
    out[(size_t)n * HH + t] = hw[(size_t)n * HH + t] * d * d;
}

// out[dst,:] += hw[src,:] * dinv[src]*dinv[dst]
__global__ __launch_bounds__(HH)
void edge_agg_kernel(const float* __restrict__ hw, const float* __restrict__ dinv,
                     const int* __restrict__ src, const int* __restrict__ dst,
                     float* __restrict__ out) {
    int e = blockIdx.x, t = threadIdx.x;
    int s = src[e], d = dst[e];
    float nrm = dinv[s] * dinv[d];
    atomicAdd(&out[(size_t)d * HH + t], hw[(size_t)s * HH + t] * nrm);
}

__global__ void bias_act_kernel(float* __restrict__ x, const float* __restrict__ b,
                                int total, int relu) {
    int i = blockIdx.x * blockDim.x + threadIdx.x;
    if (i >= total) return;
    float v = x[i] + b[i & (HH - 1)];
    x[i] = (relu && v < 0.f) ? 0.f : v;
}

__global__ __launch_bounds__(HH)
void wx_kernel(const float* __restrict__ x, const float* __restrict__ sw,
               float* __restrict__ wx) {
    int n = blockIdx.x, t = threadIdx.x;
    wx[(size_t)n * HH + t] = x[(size_t)n * HH + t] * sw[n];
}

__global__ __launch_bounds__(HH)
void sto_scatter_kernel(const float* __restrict__ wx, const int* __restrict__ batch,
                        float* __restrict__ sto) {
    int n = blockIdx.x, t = threadIdx.x;
    atomicAdd(&sto[(size_t)batch[n] * HH + t], wx[(size_t)n * HH + t]);
}

__global__ void lstm_pw_kernel(const float* __restrict__ gates,
                               float* __restrict__ c, float* __restrict__ h) {
    int i = blockIdx.x * blockDim.x + threadIdx.x;
    if (i >= BB * HH) return;
    int b = i >> 8, d = i & (HH - 1);
    const float* g = gates + (size_t)b * 4 * HH;
    float ig = g[d], fg = g[HH + d], gg = g[2 * HH + d], og = g[3 * HH + d];
    float si = 1.f / (1.f + expf(-ig));
    float sf = 1.f / (1.f + expf(-fg));
    float so = 1.f / (1.f + expf(-og));
    float cn = sf * c[i] + si * tanhf(gg);
    c[i] = cn;
    h[i] = so * tanhf(cn);
}

// e[n] = dot(wx[n,:], h[batch[n],:])  -- one wave per node
__global__ __launch_bounds__(256)
void attn_e_kernel(const float* __restrict__ wx, const float* __restrict__ h,
                   const int* __restrict__ batch, float* __restrict__ e) {
    int wave = threadIdx.x >> 5, lane = threadIdx.x & 31;
    int n = blockIdx.x * 8 + wave;
    if (n >= NN) return;
    const float* wr = wx + (size_t)n * HH;
    const float* hr = h + (size_t)batch[n] * HH;
    float s = 0.f;
    for (int d = lane; d < HH; d += 32) s += wr[d] * hr[d];
    for (int m = 16; m; m >>= 1) s += __shfl_xor(s, m, 32);
    if (lane == 0) e[n] = s;
}

__device__ inline void atomicMaxFloat(float* addr, float val) {
    unsigned int* ua = (unsigned int*)addr;
    unsigned int old = *ua;
    while (__uint_as_float(old) < val) {
        unsigned int assumed = old;
        old = atomicCAS(ua, assumed, __float_as_uint(val));
        if (old == assumed) break;
    }
}

__global__ void seg_max_kernel(const float* __restrict__ e, const int* __restrict__ batch,
                               float* __restrict__ emax) {
    int n = blockIdx.x * blockDim.x + threadIdx.x;
    if (n < NN) atomicMaxFloat(&emax[batch[n]], e[n]);
}

__global__ void attn_norm_kernel(const float* __restrict__ e, const float* __restrict__ emax,
                                 const int* __restrict__ batch,
                                 float* __restrict__ a, float* __restrict__ denom) {
    int n = blockIdx.x * blockDim.x + threadIdx.x;
    if (n >= NN) return;
    int b = batch[n];
    float av = expf(e[n] - emax[b]);
    a[n] = av;
    atomicAdd(&denom[b], av);
}

__global__ __launch_bounds__(HH)
void r_scatter_kernel(const float* __restrict__ wx, const float* __restrict__ a,
                      const float* __restrict__ denom, const int* __restrict__ batch,
                      float* __restrict__ r) {
    int n = blockIdx.x, t = threadIdx.x;
    int b = batch[n];
    float coeff = a[n] / (denom[b] + 1e-16f);
    atomicAdd(&r[(size_t)b * HH + t], wx[(size_t)n * HH + t] * coeff);
}

__global__ void concat_kernel(const float* __restrict__ h, const float* __restrict__ r,
                              float* __restrict__ q) {
    int i = blockIdx.x * blockDim.x + threadIdx.x;
    if (i >= BB * 2 * HH) return;
    int b = i >> 9, d = i & (2 * HH - 1);
    q[i] = (d < HH) ? h[(size_t)b * HH + d] : r[(size_t)b * HH + d - HH];
}

__global__ __launch_bounds__(256)
void rownorm_kernel(float* __restrict__ m) {
    int wave = threadIdx.x >> 5, lane = threadIdx.x & 31;
    int b = blockIdx.x * 8 + wave;
    float* row = m + (size_t)b * HH;
    float s = 0.f;
    for (int d = lane; d < HH; d += 32) { float v = row[d]; s += v * v; }
    for (int mm = 16; mm; mm >>= 1) s += __shfl_xor(s, mm, 32);
    float inv = 1.0f / fmaxf(sqrtf(s), 1e-12f);
    for (int d = lane; d < HH; d += 32) row[d] *= inv;
}

__global__ __launch_bounds__(256)
void dot_p2_kernel(const float* __restrict__ y, const float* __restrict__ w,
                   const float* __restrict__ b, float* __restrict__ out) {
    int wave = threadIdx.x >> 5, lane = threadIdx.x & 31;
    int r = blockIdx.x * 8 + wave;
    const float* row = y + (size_t)r * 128;
    float s = 0.f;
    for (int d = lane; d < 128; d += 32) s += row[d] * w[d];
    for (int m = 16; m; m >>= 1) s += __shfl_xor(s, m, 32);
    if (lane == 0) out[r] = s + b[0];
}

// ---------------------------------------------------------------------------
extern "C" void kernel_launch(void* const* d_in, const int* in_sizes, int n_in,
                              void* d_out, int out_size, void* d_ws, size_t ws_size,
                              hipStream_t stream) {
    const float* weight   = (const float*)d_in[0];
    const int*   sto_x    = (const int*)  d_in[1];
    const int*   e_src    = (const int*)  d_in[2];          // [2,E] row 0
    const int*   e_dst    = ((const int*) d_in[2]) + EE;    // row 1
    const float* sto_w    = (const float*)d_in[3];
    const int*   batch    = (const int*)  d_in[4];
    const float* gW0      = (const float*)d_in[5];
    const float* gb0      = (const float*)d_in[6];
    const float* gW1      = (const float*)d_in[7];
    const float* gb1      = (const float*)d_in[8];
    const float* gW2      = (const float*)d_in[9];
    const float* gb2      = (const float*)d_in[10];
    const float* Wih      = (const float*)d_in[11];   // [1024,512]
    const float* Whh      = (const float*)d_in[12];   // [1024,256]
    const float* bih      = (const float*)d_in[13];
    const float* bhh      = (const float*)d_in[14];
    const float* mlinW    = (const float*)d_in[15];   // [256,512]
    const float* mlinb    = (const float*)d_in[16];
    const float* p0W      = (const float*)d_in[17];   // [128,256]
    const float* p0b      = (const float*)d_in[18];
    const float* p1W      = (const float*)d_in[19];   // [128,128]
    const float* p1b      = (const float*)d_in[20];
    const float* p2W      = (const float*)d_in[21];   // [1,128]
    const float* p2b      = (const float*)d_in[22];
    float* out = (float*)d_out;

    float* ws = (float*)d_ws;
    size_t off = 0;
    auto alloc = [&](size_t n) { float* p = ws + off; off += n; return p; };
    float* xe    = alloc((size_t)NN * DIN);
    float* hw    = alloc((size_t)NN * HH);
    float* hA    = alloc((size_t)NN * HH);
    float* hB    = alloc((size_t)NN * HH);
    float* dinv  = alloc(NN);
    float* eN    = alloc(NN);
    float* aN    = alloc(NN);
    float* sto   = alloc((size_t)BB * HH);
    float* hS    = alloc((size_t)BB * HH);
    float* cS    = alloc((size_t)BB * HH);
    float* rS    = alloc((size_t)BB * HH);
    float* qstar = alloc((size_t)BB * 2 * HH);
    float* gates = alloc((size_t)BB * 4 * HH);
    float* meanB = alloc((size_t)BB * HH);
    float* y0    = alloc((size_t)BB * 128);
    float* y1    = alloc((size_t)BB * 128);
    float* emax  = alloc(BB);
    float* denom = alloc(BB);
    (void)ws_size; (void)n_in; (void)in_sizes; (void)out_size;

    auto grid_of = [](int M, int Nout) { return dim3(M / 16, Nout / 128); };

    // ---- degrees & symmetric norm ----
    fill_kernel<<<(NN + 255) / 256, 256, 0, stream>>>(dinv, 1.0f, NN); // self loops
    deg_kernel<<<(EE + 255) / 256, 256, 0, stream>>>(e_dst, dinv, EE);
    dinv_kernel<<<(NN + 255) / 256, 256, 0, stream>>>(dinv, NN);

    // ---- embedding gather ----
    embed_kernel<<<((size_t)NN * DIN + 255) / 256, 256, 0, stream>>>(weight, sto_x, xe,
                                                                     NN * DIN);

    // ---- GCN layer 0 ----
    wmma_gemm<false, 0, false><<<grid_of(NN, HH), 256, 0, stream>>>(
        xe, gW0, nullptr, hw, NN, DIN, HH);
    self_init_kernel<<<NN, HH, 0, stream>>>(hw, dinv, hA);
    edge_agg_kernel<<<EE, HH, 0, stream>>>(hw, dinv, e_src, e_dst, hA);
    bias_act_kernel<<<((size_t)NN * HH + 255) / 256, 256, 0, stream>>>(hA, gb0, NN * HH, 1);
    // ---- GCN layer 1 ----
    wmma_gemm<false, 0, false><<<grid_of(NN, HH), 256, 0, stream>>>(
        hA, gW1, nullptr, hw, NN, HH, HH);
    self_init_kernel<<<NN, HH, 0, stream>>>(hw, dinv, hB);
    edge_agg_kernel<<<EE, HH, 0, stream>>>(hw, dinv, e_src, e_dst, hB);
    bias_act_kernel<<<((size_t)NN * HH + 255) / 256, 256, 0, stream>>>(hB, gb1, NN * HH, 1);
    // ---- GCN layer 2 (no relu) ----
    wmma_gemm<false, 0, false><<<grid_of(NN, HH), 256, 0, stream>>>(
        hB, gW2, nullptr, hw, NN, HH, HH);
    self_init_kernel<<<NN, HH, 0, stream>>>(hw, dinv, hA);
    edge_agg_kernel<<<EE, HH, 0, stream>>>(hw, dinv, e_src, e_dst, hA);
    bias_act_kernel<<<((size_t)NN * HH + 255) / 256, 256, 0, stream>>>(hA, gb2, NN * HH, 0);

    // ---- stoichiometric weighting + graph scatter sum ----
    wx_kernel<<<NN, HH, 0, stream>>>(hA, sto_w, hB);   // wx = hB
    hipMemsetAsync(sto, 0, (size_t)BB * HH * sizeof(float), stream);
    sto_scatter_kernel<<<NN, HH, 0, stream>>>(hB, batch, sto);

    // ---- Set2Set (2 steps) ----
    hipMemsetAsync(hS, 0, (size_t)BB * HH * sizeof(float), stream);
    hipMemsetAsync(cS, 0, (size_t)BB * HH * sizeof(float), stream);
    hipMemsetAsync(qstar, 0, (size_t)BB * 2 * HH * sizeof(float), stream);
    for (int step = 0; step < 2; ++step) {
        wmma_gemm<true, 0, false><<<grid_of(BB, 4 * HH), 256, 0, stream>>>(
            qstar, Wih, bih, gates, BB, 2 * HH, 4 * HH);
        wmma_gemm<true, 0, true><<<grid_of(BB, 4 * HH), 256, 0, stream>>>(
            hS, Whh, bhh, gates, BB, HH, 4 * HH);
        lstm_pw_kernel<<<((size_t)BB * HH + 255) / 256, 256, 0, stream>>>(gates, cS, hS);
        attn_e_kernel<<<(NN + 7) / 8, 256, 0, stream>>>(hB, hS, batch, eN);
        fill_kernel<<<(BB + 255) / 256, 256, 0, stream>>>(emax, -INFINITY, BB);
        seg_max_kernel<<<(NN + 255) / 256, 256, 0, stream>>>(eN, batch, emax);
        hipMemsetAsync(denom, 0, BB * sizeof(float), stream);
        attn_norm_kernel<<<(NN + 255) / 256, 256, 0, stream>>>(eN, emax, batch, aN, denom);
        hipMemsetAsync(rS, 0, (size_t)BB * HH * sizeof(float), stream);
        r_scatter_kernel<<<NN, HH, 0, stream>>>(hB, aN, denom, batch, rS);
        concat_kernel<<<((size_t)BB * 2 * HH + 255) / 256, 256, 0, stream>>>(hS, rS, qstar);
    }

    // ---- mean = q_star @ mlin_W.T + mlin_b + sto ; row-normalize ----
    hipMemcpyAsync(meanB, sto, (size_t)BB * HH * sizeof(float),
                   hipMemcpyDeviceToDevice, stream);
    wmma_gemm<true, 0, true><<<grid_of(BB, HH), 256, 0, stream>>>(
        qstar, mlinW, mlinb, meanB, BB, 2 * HH, HH);
    rownorm_kernel<<<BB / 8, 256, 0, stream>>>(meanB);

    // ---- predictor MLP ----
    wmma_gemm<true, 2, false><<<grid_of(BB, 128), 256, 0, stream>>>(
        meanB, p0W, p0b, y0, BB, HH, 128);
    wmma_gemm<true, 2, false><<<grid_of(BB, 128), 256, 0, stream>>>(
        y0, p1W, p1b, y1, BB, 128, 128);
    dot_p2_kernel<<<BB / 8, 256, 0, stream>>>(y1, p2W, p2b, out);
}